// SeriesDecompEMA_25211458028033
// MI455X (gfx1250) — compile-verified
//
#include <hip/hip_runtime.h>
#include <hip/hip_bf16.h>

typedef __attribute__((ext_vector_type(2))) float v2f;
typedef __attribute__((ext_vector_type(8))) float v8f;

// Problem constants (from reference): x[B,T,C] f32, alpha scalar f32.
#define BB 64
#define TT 720
#define CC 512
#define CHUNK 16
#define NCHUNK (TT / CHUNK)      // 45
#define CTILES (CC / 16)         // 32 channel tiles
#define WAVES_PER_BLOCK 8

// One wave32 per (batch, 16-channel tile). Sequential scan over 45 time
// chunks; each chunk is D = A(16x16 lower-tri, f32) * Y(16x16) + C(carry),
// realized as 4 chained V_WMMA_F32_16X16X4_F32.
__global__ __launch_bounds__(256) void ema_decomp_wmma_kernel(
    const float* __restrict__ x, const float* __restrict__ alpha_p,
    float* __restrict__ out)
{
    const int lane = threadIdx.x & 31;
    const int wave = blockIdx.x * WAVES_PER_BLOCK + (threadIdx.x >> 5);
    const int b  = wave >> 5;        // wave / CTILES  (CTILES == 32)
    const int ct = wave & 31;        // wave % CTILES
    const int cb = ct * 16;

    const int hi = lane >> 4;        // 0: lanes 0-15, 1: lanes 16-31
    const int ln = lane & 15;

    const float alpha = alpha_p[0];
    const float om = 1.0f - alpha;

    // om^0 .. om^15 (per-lane scalar chain, computed once)
    float omp[16];
    omp[0] = 1.0f;
#pragma unroll
    for (int i = 1; i < 16; ++i) omp[i] = omp[i - 1] * om;

    // A fragments (f32 16x4 layout): lane -> M = lane&15;
    // VGPR j -> K_local = j + 2*hi; global col c = 4*kb + K_local.
    // A[M][c] = (M >= c) ? om^(M-c) : 0   (alpha folded into B operand)
    v2f a[4];
#pragma unroll
    for (int kb = 0; kb < 4; ++kb) {
#pragma unroll
        for (int j = 0; j < 2; ++j) {
            const int c = 4 * kb + j + 2 * hi;
            a[kb][j] = (ln >= c) ? omp[ln - c] : 0.0f;
        }
    }

    // Carry decay per C/D VGPR v: row M = v + 8*hi, factor om^(M+1)
    float dp[8];
    {
        float p = om * (hi ? omp[8] : 1.0f);   // om^(1 + 8*hi)
#pragma unroll
        for (int v = 0; v < 8; ++v) { dp[v] = p; p *= om; }
    }

    const float* __restrict__ xb = x + (size_t)b * TT * CC;
    float* __restrict__ res_out = out;                              // output 0: res
    float* __restrict__ ma_out  = out + (size_t)BB * TT * CC;       // output 1: ma

    float carry = 0.0f;   // ma at (t0-1) for channel cb+ln (replicated in both halves)

    for (int ch = 0; ch < NCHUNK; ++ch) {
        const int t0 = ch * CHUNK;

        // Prefetch next chunk's tile (global_prefetch_b8); harmless on last iter.
        if (ch + 1 < NCHUNK) {
            __builtin_prefetch(&xb[(size_t)(t0 + CHUNK + lane % 16) * CC + cb], 0, 1);
        }

        // C accumulator init: rank-1 carry term  acc[v] = om^(M+1) * carry
        v8f acc;
#pragma unroll
        for (int v = 0; v < 8; ++v) acc[v] = dp[v] * carry;

        // 4 chained K=4 WMMAs: D = sum_kb A[kb] x B[kb] + C
#pragma unroll
        for (int kb = 0; kb < 4; ++kb) {
            v2f bfrag;
#pragma unroll
            for (int j = 0; j < 2; ++j) {
                const int k = 4 * kb + j + 2 * hi;   // B row (time within chunk)
                const int t = t0 + k;
                const float xv = xb[(size_t)t * CC + cb + ln];
                const float s  = (t == 0) ? 1.0f : alpha;  // y = alpha*x, except t==0
                bfrag[j] = s * xv;
            }
            acc = __builtin_amdgcn_wmma_f32_16x16x4_f32(
                false, a[kb], false, bfrag, (short)0, acc, false, false);
        }

        // Next carry: D row 15 lives in VGPR 7 of lanes 16..31 (N = lane-16).
        carry = __shfl(acc[7], 16 + ln, 32);

        // Store ma and res. D layout: VGPR v, this lane -> M = v + 8*hi, N = ln.
#pragma unroll
        for (int v = 0; v < 8; ++v) {
            const int t = t0 + v + 8 * hi;
            const size_t idx = (size_t)b * TT * CC + (size_t)t * CC + cb + ln;
            const float ma = acc[v];
            const float xv = xb[(size_t)t * CC + cb + ln];   // WGP$/L2 hit
            ma_out[idx]  = ma;
            res_out[idx] = xv - ma;
        }
    }
}

extern "C" void kernel_launch(void* const* d_in, const int* in_sizes, int n_in,
                              void* d_out, int out_size, void* d_ws, size_t ws_size,
                              hipStream_t stream) {
    (void)in_sizes; (void)n_in; (void)d_ws; (void)ws_size; (void)out_size;
    const float* x     = (const float*)d_in[0];
    const float* alpha = (const float*)d_in[1];
    float* out         = (float*)d_out;

    const int total_waves = BB * CTILES;                 // 2048
    const int blocks = total_waves / WAVES_PER_BLOCK;    // 256
    ema_decomp_wmma_kernel<<<blocks, WAVES_PER_BLOCK * 32, 0, stream>>>(x, alpha, out);
}